// HybridAugmentor_7619271983308
// MI455X (gfx1250) — compile-verified
//
#include <hip/hip_runtime.h>
#include <hip/hip_bf16.h>
#include <cmath>

typedef __attribute__((ext_vector_type(2))) float v2f;
typedef __attribute__((ext_vector_type(8))) float v8f;

#define NPTS 9
#define EPS_TPS 1e-8f

// ---------------------------------------------------------------------------
// Kernel 1a: class-aware mixup, vectorized 4 pixels/thread (b128 traffic).
// mixed = bg*x + fg*(lam*x + (1-lam)*x[perm]),  bg=(sum_c m==0), fg=count(m==1)
// ---------------------------------------------------------------------------
__global__ void mixup_kernel_v4(const float* __restrict__ x,
                                const int* __restrict__ masks,
                                const float* __restrict__ lam_p,
                                const int* __restrict__ perm,
                                float* __restrict__ mixed,
                                int B, int HW) {
    const int HW4 = HW >> 2;
    long long idx4 = (long long)blockIdx.x * blockDim.x + threadIdx.x;
    long long total4 = (long long)B * HW4;
    if (idx4 >= total4) return;
    int b  = (int)(idx4 / HW4);
    int p4 = (int)(idx4 % HW4);

    const int4* mb = (const int4*)(masks + ((long long)b * 5) * HW) + p4;
    int4 ssum = make_int4(0, 0, 0, 0);
    int4 s1   = make_int4(0, 0, 0, 0);
#pragma unroll
    for (int c = 0; c < 5; ++c) {
        int4 m = mb[(long long)c * HW4];
        ssum.x += m.x; ssum.y += m.y; ssum.z += m.z; ssum.w += m.w;
        s1.x += (m.x == 1); s1.y += (m.y == 1); s1.z += (m.z == 1); s1.w += (m.w == 1);
    }
    float4 xv = ((const float4*)(x + (long long)b * HW))[p4];
    float4 xp = ((const float4*)(x + (long long)perm[b] * HW))[p4];
    float l = lam_p[0];

    float4 o;
    o.x = ((ssum.x == 0) ? 1.0f : 0.0f) * xv.x + (float)s1.x * (l * xv.x + (1.0f - l) * xp.x);
    o.y = ((ssum.y == 0) ? 1.0f : 0.0f) * xv.y + (float)s1.y * (l * xv.y + (1.0f - l) * xp.y);
    o.z = ((ssum.z == 0) ? 1.0f : 0.0f) * xv.z + (float)s1.z * (l * xv.z + (1.0f - l) * xp.z);
    o.w = ((ssum.w == 0) ? 1.0f : 0.0f) * xv.w + (float)s1.w * (l * xv.w + (1.0f - l) * xp.w);
    ((float4*)(mixed + (long long)b * HW))[p4] = o;
}

// Kernel 1b: scalar fallback (HW not divisible by 4)
__global__ void mixup_kernel(const float* __restrict__ x,
                             const int* __restrict__ masks,
                             const float* __restrict__ lam_p,
                             const int* __restrict__ perm,
                             float* __restrict__ mixed,
                             int B, int HW) {
    long long idx = (long long)blockIdx.x * blockDim.x + threadIdx.x;
    long long total = (long long)B * HW;
    if (idx >= total) return;
    int b = (int)(idx / HW);
    int p = (int)(idx % HW);
    const int* mb = masks + ((long long)b * 5) * HW + p;
    int ssum = 0, s1 = 0;
#pragma unroll
    for (int c = 0; c < 5; ++c) {
        int m = mb[(long long)c * HW];
        ssum += m;
        s1 += (m == 1) ? 1 : 0;
    }
    float bg = (ssum == 0) ? 1.0f : 0.0f;
    float fg = (float)s1;
    float xv = x[idx];
    float xp = x[(long long)perm[b] * HW + p];
    float l = lam_p[0];
    mixed[idx] = bg * xv + fg * (l * xv + (1.0f - l) * xp);
}

// ---------------------------------------------------------------------------
// Kernel 2: per-batch TPS fit. One thread per batch element (128 total).
// Elasticity from 5x5 mask regions -> displaced control points -> Gauss-Jordan
// with partial pivoting (L's diagonal is all zeros), both noise sets at once.
// ---------------------------------------------------------------------------
__global__ void tps_solve_kernel(const int* __restrict__ masks,
                                 const float* __restrict__ noise_g,
                                 const float* __restrict__ noise_l,
                                 float* __restrict__ wg,   // [B][12][2]
                                 float* __restrict__ wl,   // [B][12][2]
                                 int B, int H, int W) {
    int b = blockIdx.x * blockDim.x + threadIdx.x;
    if (b >= B) return;

    const float EL[5] = {0.05f, 0.12f, 0.08f, 0.15f, 0.10f};
    float elas[NPTS];
#pragma unroll
    for (int n = 0; n < NPTS; ++n) {
        int i = n / 3, j = n % 3;
        int yc = (int)((i * 0.5f) * (H - 1));   // python int() truncation
        int xc = (int)((j * 0.5f) * (W - 1));
        int y0 = max(0, yc - 2), y1 = min(H, yc + 3);
        int x0 = max(0, xc - 2), x1 = min(W, xc + 3);
        int s[5] = {0, 0, 0, 0, 0};
        for (int c = 0; c < 5; ++c) {
            const int* mp = masks + (((long long)b * 5 + c) * H) * W;
            for (int yy = y0; yy < y1; ++yy)
                for (int xx = x0; xx < x1; ++xx)
                    s[c] += mp[(long long)yy * W + xx];
        }
        int dom = 0;
        for (int c = 1; c < 5; ++c) if (s[c] > s[dom]) dom = c;   // first-max
        elas[n] = EL[dom];
    }

    float dg[NPTS][2], dl[NPTS][2];
#pragma unroll
    for (int n = 0; n < NPTS; ++n) {
        dg[n][0] = noise_g[((long long)b * NPTS + n) * 2 + 0] * elas[n];
        dg[n][1] = noise_g[((long long)b * NPTS + n) * 2 + 1] * elas[n];
        dl[n][0] = noise_l[((long long)b * NPTS + n) * 2 + 0] * elas[n];
        dl[n][1] = noise_l[((long long)b * NPTS + n) * 2 + 1] * elas[n];
    }
    // center blend: d4 = 0.8*d4 + 0.2*0.25*(d1+d7+d3+d5)
#pragma unroll
    for (int d = 0; d < 2; ++d) {
        dg[4][d] = 0.8f * dg[4][d] + 0.05f * (dg[1][d] + dg[7][d] + dg[3][d] + dg[5][d]);
        dl[4][d] = 0.8f * dl[4][d] + 0.05f * (dl[1][d] + dl[7][d] + dl[3][d] + dl[5][d]);
    }

    // control grid: src0 = i-1, src1 = j-1  (values in {-1,0,1})
    float c0[NPTS], c1[NPTS];
#pragma unroll
    for (int n = 0; n < NPTS; ++n) { c0[n] = (float)(n / 3) - 1.0f; c1[n] = (float)(n % 3) - 1.0f; }

    // augmented system M = [ L | rhs_g | rhs_l ]  (12 x 16)
    float M[12][16];
    for (int a = 0; a < NPTS; ++a) {
        for (int q = 0; q < NPTS; ++q) {
            float dx = c0[a] - c0[q], dy = c1[a] - c1[q];
            float r2 = dx * dx + dy * dy;
            M[a][q] = 0.5f * r2 * __logf(r2 + EPS_TPS);
        }
        M[a][9] = 1.0f; M[a][10] = c0[a]; M[a][11] = c1[a];
        M[a][12] = c0[a] + dg[a][0];
        M[a][13] = c1[a] + dg[a][1];
        M[a][14] = c0[a] + dl[a][0];
        M[a][15] = c1[a] + dl[a][1];
    }
    for (int r = 9; r < 12; ++r) {
        for (int q = 0; q < NPTS; ++q)
            M[r][q] = (r == 9) ? 1.0f : (r == 10 ? c0[q] : c1[q]);
        for (int q = 9; q < 16; ++q) M[r][q] = 0.0f;
    }

    // Gauss-Jordan with partial pivoting
    for (int k = 0; k < 12; ++k) {
        int piv = k; float best = fabsf(M[k][k]);
        for (int r = k + 1; r < 12; ++r) {
            float v = fabsf(M[r][k]);
            if (v > best) { best = v; piv = r; }
        }
        if (piv != k)
            for (int q = 0; q < 16; ++q) { float t = M[k][q]; M[k][q] = M[piv][q]; M[piv][q] = t; }
        float invp = 1.0f / M[k][k];
        for (int r = 0; r < 12; ++r) {
            if (r == k) continue;
            float fac = M[r][k] * invp;
            for (int q = k; q < 16; ++q) M[r][q] -= fac * M[k][q];
        }
    }
    for (int k = 0; k < 12; ++k) {
        float invp = 1.0f / M[k][k];
        wg[(long long)b * 24 + 2 * k + 0] = M[k][12] * invp;
        wg[(long long)b * 24 + 2 * k + 1] = M[k][13] * invp;
        wl[(long long)b * 24 + 2 * k + 0] = M[k][14] * invp;
        wl[(long long)b * 24 + 2 * k + 1] = M[k][15] * invp;
    }
}

// ---------------------------------------------------------------------------
// Kernel 3: TPS warp + bilinear sample.
// Wave handles 16 pixels; [12-feature] x [12x2 weight] coordinate transform
// runs as 3 chained V_WMMA_F32_16X16X4_F32 with pixels on the N axis:
// lane n (0..15) receives warped-x in D vgpr0 (M=0) and warped-y in vgpr1.
//
// Codegen hygiene:
//  * all 12 features are NAMED SCALARS (no array -> no select-of-GEP ->
//    no dynamic-extract cndmask trees); each half-select is one v_cndmask
//  * weight loads are unconditional in-bounds lane-addressed loads
//  * bilinear validity uses eager bitwise & predicates + unconditional
//    clamped loads + one v_cndmask per corner (no saveexec nests)
// ---------------------------------------------------------------------------
__global__ void tps_warp_kernel(const float* __restrict__ img,   // [B][H*W]
                                const float* __restrict__ wts,   // [B][12][2]
                                float* __restrict__ out,         // [B][H*W]
                                int B, int H, int W) {
    const int HW = H * W;
    const int b = blockIdx.y;
    const int lane = threadIdx.x & 31;
    const int m = lane & 15;          // pixel-in-wave / matrix N index
    const bool hi = lane >= 16;       // upper half supplies K=+2,+3

    const int wave = threadIdx.x >> 5;
    const int p = blockIdx.x * 128 + wave * 16 + m;   // pixel id
    const int pc = (p < HW) ? p : (HW - 1);           // clamp; store is guarded

    const int px = pc % W;
    const int py = pc / W;
    const float Wm1 = (float)(W - 1), Hm1 = (float)(H - 1);
    const float ptx = 2.0f * (float)px / Wm1 - 1.0f;
    const float pty = 2.0f * (float)py / Hm1 - 1.0f;

    // --- 12 features as named scalars (no arrays!) ---
    // centers: c0 = i-1 pairs with ptx, c1 = j-1 pairs with pty, n = i*3+j
    const float ax0 = ptx + 1.0f, ax1 = ptx, ax2 = ptx - 1.0f;
    const float ay0 = pty + 1.0f, ay1 = pty, ay2 = pty - 1.0f;
    const float sx0 = ax0 * ax0, sx1 = ax1 * ax1, sx2 = ax2 * ax2;
    const float sy0 = ay0 * ay0, sy1 = ay1 * ay1, sy2 = ay2 * ay2;
#define TPSU(r2) (0.5f * (r2) * __logf((r2) + EPS_TPS))
    const float k0 = TPSU(sx0 + sy0);   // n=0: i=0,j=0
    const float k1 = TPSU(sx0 + sy1);   // n=1: i=0,j=1
    const float k2 = TPSU(sx0 + sy2);   // n=2: i=0,j=2
    const float k3 = TPSU(sx1 + sy0);   // n=3
    const float k4 = TPSU(sx1 + sy1);   // n=4
    const float k5 = TPSU(sx1 + sy2);   // n=5
    const float k6 = TPSU(sx2 + sy0);   // n=6
    const float k7 = TPSU(sx2 + sy1);   // n=7
    const float k8 = TPSU(sx2 + sy2);   // n=8
#undef TPSU
    // f9 = 1, f10 = ptx, f11 = pty

    const float* Wb = wts + (long long)b * 24;   // [k][2]: col0 = x, col1 = y
    const int col = m & 1;                        // weight column this lane may need
    const int kh  = hi ? 2 : 0;                   // K offset for this half
    const float keepA = (m < 2) ? 1.0f : 0.0f;    // rows M>=2 of A are zero

    // unconditional in-bounds weight loads (one cacheline per wave per step)
    const float wA0 = Wb[2 * kh + col];           // K = kh
    const float wA1 = Wb[2 * kh + 2 + col];       // K = kh+1
    const float wB0 = Wb[8 + 2 * kh + col];       // K = 4+kh
    const float wB1 = Wb[8 + 2 * kh + 2 + col];   // K = 4+kh+1
    const float wC0 = Wb[16 + 2 * kh + col];      // K = 8+kh
    const float wC1 = Wb[16 + 2 * kh + 2 + col];  // K = 8+kh+1

    v8f acc = {};
    {   // step 0: K = 0..3
        v2f A, Bv;
        A.x = keepA * wA0;  A.y = keepA * wA1;
        Bv.x = hi ? k2 : k0;
        Bv.y = hi ? k3 : k1;
        acc = __builtin_amdgcn_wmma_f32_16x16x4_f32(false, A, false, Bv, (short)0, acc, false, false);
    }
    {   // step 1: K = 4..7
        v2f A, Bv;
        A.x = keepA * wB0;  A.y = keepA * wB1;
        Bv.x = hi ? k6 : k4;
        Bv.y = hi ? k7 : k5;
        acc = __builtin_amdgcn_wmma_f32_16x16x4_f32(false, A, false, Bv, (short)0, acc, false, false);
    }
    {   // step 2: K = 8..11  (f = k8, 1, ptx, pty)
        v2f A, Bv;
        A.x = keepA * wC0;  A.y = keepA * wC1;
        Bv.x = hi ? ptx : k8;
        Bv.y = hi ? pty : 1.0f;
        acc = __builtin_amdgcn_wmma_f32_16x16x4_f32(false, A, false, Bv, (short)0, acc, false, false);
    }

    // lanes 0-15: D[0][n] = warped x of pixel n, D[1][n] = warped y
    const float wx = acc[0];
    const float wy = acc[1];

    if (!hi && p < HW) {
        const float gx = (wx + 1.0f) * 0.5f * Wm1;
        const float gy = (wy + 1.0f) * 0.5f * Hm1;
        const float x0f = floorf(gx), y0f = floorf(gy);
        const float x1f = x0f + 1.0f, y1f = y0f + 1.0f;
        const float fx1 = gx - x0f, fx0 = 1.0f - fx1;
        const float fy1 = gy - y0f, fy0 = 1.0f - fy1;

        // eager (non-short-circuit) validity predicates
        const bool vx0 = (x0f >= 0.0f) & (x0f <= Wm1);
        const bool vx1 = (x1f >= 0.0f) & (x1f <= Wm1);
        const bool vy0 = (y0f >= 0.0f) & (y0f <= Hm1);
        const bool vy1 = (y1f >= 0.0f) & (y1f <= Hm1);

        const int xi0 = (int)fminf(fmaxf(x0f, 0.0f), Wm1);
        const int xi1 = (int)fminf(fmaxf(x1f, 0.0f), Wm1);
        const int yi0 = (int)fminf(fmaxf(y0f, 0.0f), Hm1);
        const int yi1 = (int)fminf(fmaxf(y1f, 0.0f), Hm1);

        const float* ib = img + (long long)b * HW;
        float v00 = ib[(long long)yi0 * W + xi0];
        float v01 = ib[(long long)yi0 * W + xi1];
        float v10 = ib[(long long)yi1 * W + xi0];
        float v11 = ib[(long long)yi1 * W + xi1];
        v00 = (vy0 & vx0) ? v00 : 0.0f;
        v01 = (vy0 & vx1) ? v01 : 0.0f;
        v10 = (vy1 & vx0) ? v10 : 0.0f;
        v11 = (vy1 & vx1) ? v11 : 0.0f;

        out[(long long)b * HW + p] = v00 * (fy0 * fx0) + v01 * (fy0 * fx1)
                                   + v10 * (fy1 * fx0) + v11 * (fy1 * fx1);
    }
}

// ---------------------------------------------------------------------------
// inputs: x, masks, lam, perm, noise_g, noise_l, epoch, total_epochs
// outputs: (global_aug, local_aug) concatenated in d_out
// ---------------------------------------------------------------------------
extern "C" void kernel_launch(void* const* d_in, const int* in_sizes, int n_in,
                              void* d_out, int out_size, void* d_ws, size_t ws_size,
                              hipStream_t stream) {
    const float* x     = (const float*)d_in[0];
    const int*   masks = (const int*)d_in[1];
    const float* lam   = (const float*)d_in[2];
    const int*   perm  = (const int*)d_in[3];
    const float* ng    = (const float*)d_in[4];
    const float* nl    = (const float*)d_in[5];

    const int B  = in_sizes[3];               // perm has B entries
    const int HW = in_sizes[0] / B;           // x is [B,1,H,W]
    int H = (int)(sqrtf((float)HW) + 0.5f);
    int W = HW / H;

    float* mixed = (float*)d_ws;              // B*HW floats
    float* wg    = mixed + (size_t)B * HW;    // B*24 floats
    float* wl    = wg + (size_t)B * 24;       // B*24 floats

    float* out_g = (float*)d_out;
    float* out_l = out_g + (size_t)B * HW;

    if ((HW & 3) == 0) {
        long long total4 = (long long)B * (HW >> 2);
        int blocks = (int)((total4 + 255) / 256);
        mixup_kernel_v4<<<blocks, 256, 0, stream>>>(x, masks, lam, perm, mixed, B, HW);
    } else {
        long long total = (long long)B * HW;
        int blocks = (int)((total + 255) / 256);
        mixup_kernel<<<blocks, 256, 0, stream>>>(x, masks, lam, perm, mixed, B, HW);
    }
    tps_solve_kernel<<<(B + 63) / 64, 64, 0, stream>>>(masks, ng, nl, wg, wl, B, H, W);

    dim3 grid((HW + 127) / 128, B);
    tps_warp_kernel<<<grid, 256, 0, stream>>>(mixed, wg, out_g, B, H, W);
    tps_warp_kernel<<<grid, 256, 0, stream>>>(x,     wl, out_l, B, H, W);
}